// FinalLayer_multi_54881092108537
// MI455X (gfx1250) — compile-verified
//
#include <hip/hip_runtime.h>
#include <math.h>

// RBF set-conv final layer for MI455X (gfx1250, wave32).
// Shapes fixed by the harness: nb=8, ngrid=512, ntarget=1024, nbasis=5, nchannel=3.
//
// Strategy (compute-bound: ~63M exps + 63M MACs, inputs ~400KB -> L2/LDS resident):
//  - One block = 128 threads (4 waves). Block owns (b, 64-target tile).
//  - Stage x_grid[b] and h_grid[b]*g_w into LDS once.
//  - Each wave owns 16 targets; loops g in chunks of 4 using V_WMMA_F32_16X16X4_F32.
//    A tiles = Gaussian weights, one per (k,c) combo; exponent coefficient pre-folds
//              -0.5*log2(e)/s^2 so each weight is ONE v_mul + ONE raw v_exp_f32.
//              All 15 A tiles are computed into distinct registers before the WMMA
//              burst so the TRANS->consumer hazard slots are filled by other exps
//              instead of v_nop.
//    B tile  = shared across combos: B[g4, n] = h[g, kc_n]*g_w[k_n] (col 15 = 0).
//    D_kc column kc = sum_g wt_kc[t,g] * h_kc[g]  -> exactly the needed matvec.
//  - Final: extract diagonal columns to LDS, sum the 5 bases, add g_b, store.

typedef __attribute__((ext_vector_type(2))) float v2f;
typedef __attribute__((ext_vector_type(8))) float v8f;

#define NB 8
#define NG 512
#define NT 1024
#define NK 5
#define NC 3
#define KC 15  // NK*NC

#if defined(__has_builtin) && __has_builtin(__builtin_amdgcn_exp2f)
#define FAST_EXP2(x) __builtin_amdgcn_exp2f(x)   // raw v_exp_f32
#else
#define FAST_EXP2(x) exp2f(x)
#endif

__global__ __launch_bounds__(128)
void rbf_final_wmma_kernel(const float* __restrict__ x_grid,
                           const float* __restrict__ h_grid,
                           const float* __restrict__ target_x,
                           const float* __restrict__ sigma,
                           const float* __restrict__ g_w,
                           const float* __restrict__ g_b,
                           float* __restrict__ out)
{
    __shared__ float x_s[NG * 4];   // [g][c] padded to 4 floats (8 KB)
    __shared__ float h_s[NG * 16];  // [g][kc] padded to 16, pre-scaled by g_w; col 15 = 0 (32 KB)
                                    // reused after the main loop as red[wave][16][16]

    const int tid  = threadIdx.x;
    const int wave = tid >> 5;
    const int lane = tid & 31;
    const int half = lane >> 4;   // 0 -> K pair {0,1}; 1 -> K pair {2,3}
    const int ln   = lane & 15;   // A row M / B,D column N

    const int b  = blockIdx.y;
    const int t0 = blockIdx.x * 64 + wave * 16;

    const float* xg  = x_grid   + (size_t)b * NG * NC;
    const float* hg  = h_grid   + (size_t)b * NG * KC;
    const float* txp = target_x + (size_t)b * NT * NC;

    __builtin_prefetch(hg, 0, 0);
    __builtin_prefetch(txp, 0, 0);

    // ---- stage x_grid[b] into LDS (padded to 4) ----
    for (int i = tid; i < NG; i += 128) {
        x_s[i * 4 + 0] = xg[i * 3 + 0];
        x_s[i * 4 + 1] = xg[i * 3 + 1];
        x_s[i * 4 + 2] = xg[i * 3 + 2];
        x_s[i * 4 + 3] = 0.f;
    }
    // ---- stage h_grid[b]*g_w into LDS (padded to 16, col 15 zero) ----
    for (int i = tid; i < NG * 16; i += 128) {
        const int g = i >> 4, j = i & 15;
        h_s[i] = (j < KC) ? hg[g * KC + j] * g_w[j / 3] : 0.f;
    }

    // ---- per-(k,c) exponent coefficient: -0.5*log2(e) / s^2 (one v_exp per weight) ----
    float mker[KC];
#pragma unroll
    for (int kc = 0; kc < KC; ++kc) {
        const float s = __expf(sigma[kc]) + 1e-6f;
        mker[kc] = -0.72134752f / (s * s);   // 0.5 * log2(e) = 0.72134752
    }

    // ---- this wave's 16 targets: row M = ln (both lane halves use the same rows) ----
    const float tx0 = txp[(t0 + ln) * 3 + 0];
    const float tx1 = txp[(t0 + ln) * 3 + 1];
    const float tx2 = txp[(t0 + ln) * 3 + 2];

    v8f acc[KC] = {};  // 15 independent 16x16 f32 accumulators

    __syncthreads();

    // ---- main loop: 128 chunks of 4 grid points ----
    for (int g0 = 0; g0 < NG; g0 += 4) {
        const int gA = g0 + half * 2;  // this lane's K pair: gA, gA+1

        float u0[3], u1[3];
#pragma unroll
        for (int c = 0; c < 3; ++c) {
            const float t  = (c == 0) ? tx0 : ((c == 1) ? tx1 : tx2);
            const float d0 = x_s[gA * 4 + c]       - t;
            const float d1 = x_s[(gA + 1) * 4 + c] - t;
            u0[c] = d0 * d0;
            u1[c] = d1 * d1;
        }

        // B tile (shared by all 15 WMMAs): rows K = gA, gA+1 ; column n = ln
        v2f B;
        B.x = h_s[gA * 16 + ln];
        B.y = h_s[(gA + 1) * 16 + ln];

        // Phase 1: all 15 A tiles into distinct registers (30 independent v_exp_f32;
        // TRANS hazard slots are covered by the neighbouring exps, no v_nop needed).
        v2f A[KC];
#pragma unroll
        for (int kc = 0; kc < KC; ++kc) {
            const int c = kc % 3;
            A[kc].x = FAST_EXP2(u0[c] * mker[kc]);   // wt(t0+ln, gA,   kc)
            A[kc].y = FAST_EXP2(u1[c] * mker[kc]);   // wt(t0+ln, gA+1, kc)
        }

        // Phase 2: WMMA burst (15 independent accumulator chains).
#pragma unroll
        for (int kc = 0; kc < KC; ++kc) {
            acc[kc] = __builtin_amdgcn_wmma_f32_16x16x4_f32(
                false, A[kc], false, B, (short)0, acc[kc], false, false);
        }
    }

    __syncthreads();  // everyone done reading h_s -> reuse as reduction buffer

    // ---- extract diagonal column kc of each D_kc: lanes kc (M=0..7) and kc+16 (M=8..15) ----
    float* red = h_s;  // red[wave][m 0..15][kc 0..15]
#pragma unroll
    for (int kc = 0; kc < KC; ++kc) {
        if (ln == kc) {
#pragma unroll
            for (int v = 0; v < 8; ++v)
                red[wave * 256 + (half * 8 + v) * 16 + kc] = acc[kc][v];
        }
    }

    __syncthreads();

    // ---- final contraction over k (g_w already folded into B) + bias; 64 t * 3 c outputs ----
    if (tid < 192) {
        const int tl = tid / 3;
        const int c  = tid % 3;
        float s = g_b[0];
#pragma unroll
        for (int k = 0; k < NK; ++k)
            s += red[(tl >> 4) * 256 + (tl & 15) * 16 + (k * 3 + c)];
        out[((size_t)b * NT + blockIdx.x * 64 + tl) * 3 + c] = s;
    }
}

extern "C" void kernel_launch(void* const* d_in, const int* in_sizes, int n_in,
                              void* d_out, int out_size, void* d_ws, size_t ws_size,
                              hipStream_t stream) {
    const float* x_grid   = (const float*)d_in[0];
    const float* h_grid   = (const float*)d_in[1];
    const float* target_x = (const float*)d_in[2];
    const float* sigma    = (const float*)d_in[3];
    const float* g_w      = (const float*)d_in[4];
    const float* g_b      = (const float*)d_in[5];
    float* out            = (float*)d_out;

    dim3 grid(NT / 64, NB);  // 16 target-tiles x 8 batches = 128 blocks
    rbf_final_wmma_kernel<<<grid, 128, 0, stream>>>(
        x_grid, h_grid, target_x, sigma, g_w, g_b, out);
}